// DTEnergy1D_50019189129661
// MI455X (gfx1250) — compile-verified
//
#include <hip/hip_runtime.h>
#include <hip/hip_bf16.h>

#define WIDTH 384
#define NB    32
#define LEN   512
#define NITER 16

typedef __attribute__((ext_vector_type(2))) float v2f;
typedef __attribute__((ext_vector_type(8))) float v8f;

// ---------------------------------------------------------------------------
// Projection: interim = relu(conv1d(x, proj_w)), Cin=1 (tiny FLOPs -> VALU)
// ---------------------------------------------------------------------------
__global__ __launch_bounds__(256)
void proj_kernel(const float* __restrict__ x, const float* __restrict__ pw,
                 float* __restrict__ outp) {
    int idx = blockIdx.x * blockDim.x + threadIdx.x;
    if (idx >= NB * WIDTH * LEN) return;
    int l  = idx & (LEN - 1);
    int co = (idx >> 9) % WIDTH;
    int b  = idx / (WIDTH * LEN);
    float s = 0.f;
#pragma unroll
    for (int k = 0; k < 3; ++k) {
        int ll = l + k - 1;
        if (ll >= 0 && ll < LEN) s += pw[co * 3 + k] * x[b * LEN + ll];
    }
    outp[idx] = fmaxf(s, 0.f);
}

// ---------------------------------------------------------------------------
// Per-batch LayerNorm stats over (C,L): mean + rsqrt(var+eps)
// ---------------------------------------------------------------------------
__global__ __launch_bounds__(256)
void ln_stats_kernel(const float* __restrict__ in, float* __restrict__ stats) {
    __shared__ float ssum[256];
    __shared__ float ssq[256];
    const int b = blockIdx.x, tid = threadIdx.x;
    const float* p = in + (size_t)b * WIDTH * LEN;
    float s = 0.f, q = 0.f;
    for (int i = tid; i < WIDTH * LEN; i += 256) {
        float v = p[i];
        s += v; q += v * v;
    }
    ssum[tid] = s; ssq[tid] = q;
    __syncthreads();
    for (int st = 128; st > 0; st >>= 1) {
        if (tid < st) { ssum[tid] += ssum[tid + st]; ssq[tid] += ssq[tid + st]; }
        __syncthreads();
    }
    if (tid == 0) {
        const float n  = (float)(WIDTH * LEN);
        float mu  = ssum[0] / n;
        float var = ssq[0] / n - mu * mu;
        stats[b]      = mu;
        stats[NB + b] = rsqrtf(var + 1e-5f);
    }
}

// ---------------------------------------------------------------------------
// Epilogue for one 16x16 WMMA tile.
// MODE 0: energy update  (add x-channel, on-the-fly LN, gradient step)
// MODE 1: relu + store
// ---------------------------------------------------------------------------
template <int MODE>
__device__ __forceinline__
void epilogue_tile(v8f acc, int b, int co_w, int pos_w,
                   const float* __restrict__ in, const float* __restrict__ w,
                   int w_cin, int cout_total,
                   const float* __restrict__ xin, const float* __restrict__ stats,
                   float* __restrict__ outp, int lane16, int hi) {
    const int pos = pos_w + lane16;
#pragma unroll
    for (int j = 0; j < 8; ++j) {
        const int co = co_w + j + (hi ? 8 : 0);
        float c = acc[j];
        if (MODE == 0) {
            // extra input channel ci = w_cin-1 (= raw x)
#pragma unroll
            for (int k = 0; k < 3; ++k) {
                int l = pos + k - 1;
                if (l >= 0 && l < LEN)
                    c += w[(size_t)co * w_cin * 3 + (w_cin - 1) * 3 + k] *
                         xin[b * LEN + l];
            }
            float mu   = stats[b];
            float rstd = stats[NB + b];
            float araw = in[((size_t)b * WIDTH + co) * LEN + pos];
            float xl   = (araw - mu) * rstd;
            float g    = c + 2.f * fmaxf(xl, 0.f);
            outp[((size_t)b * WIDTH + co) * LEN + pos] = xl - 0.1f * g;
        } else {
            outp[((size_t)b * cout_total + co) * LEN + pos] = fmaxf(c, 0.f);
        }
    }
}

// ---------------------------------------------------------------------------
// K=3 conv as WMMA f32 GEMM.  256 threads = 8 waves (2 x 4); block tile
// 64(cout) x 128(pos); each wave owns a 2x2 grid of 16x16
// v_wmma_f32_16x16x4_f32 accumulators (32x32 output) -> 2 A-frags + 2 B-frags
// feed 4 WMMAs, halving LDS bytes per WMMA vs a 1-tile wave.
// ---------------------------------------------------------------------------
template <int MODE>
__global__ __launch_bounds__(256)
void conv_gemm_kernel(const float* __restrict__ in,    // [B, 384, L]
                      const float* __restrict__ w,     // [Cout, w_cin, 3]
                      int w_cin, int cout_total,
                      const float* __restrict__ xin,   // MODE0: x [B,1,L]
                      const float* __restrict__ stats, // MODE0: mean/rstd
                      float* __restrict__ outp) {
    // [k][cout_local][cin_local]: lane A operand = contiguous b64
    __shared__ __align__(16) float Ws[3][64][18];
    // [pos_local(l0-1 .. l0+128)][cin_local]: lane B operand = contiguous b64
    __shared__ __align__(16) float Xs[130][18];

    const int tid    = threadIdx.x;
    const int b      = blockIdx.x >> 2;
    const int l0     = (blockIdx.x & 3) << 7;
    const int co0    = blockIdx.y << 6;
    const int wv     = tid >> 5;
    const int lane   = tid & 31;
    const int mg     = wv >> 2;        // 0..1 : wave cout offset mg*32
    const int ng     = wv & 3;         // 0..3 : wave pos  offset ng*32
    const int lane16 = lane & 15;
    const int hi     = lane >> 4;      // 0: K0/K1 half, 1: K2/K3 half

    v8f acc00 = {0.f, 0.f, 0.f, 0.f, 0.f, 0.f, 0.f, 0.f};
    v8f acc01 = acc00, acc10 = acc00, acc11 = acc00;

    for (int cc = 0; cc < WIDTH; cc += 16) {
        // stage weights: 3*64*16 elements
        for (int e = tid; e < 3 * 64 * 16; e += 256) {
            int k   = e / (64 * 16);
            int r   = e - k * 64 * 16;
            int col = r >> 4;
            int cil = r & 15;
            Ws[k][col][cil] = w[(size_t)(co0 + col) * w_cin * 3 + (cc + cil) * 3 + k];
        }
        // stage activations with halo (zero pad at batch edges): 130*16 elements
        for (int e = tid; e < 130 * 16; e += 256) {
            int p   = e >> 4;
            int cil = e & 15;
            int l   = l0 - 1 + p;
            float v = 0.f;
            if (l >= 0 && l < LEN)
                v = in[((size_t)b * WIDTH + cc + cil) * LEN + l];
            Xs[p][cil] = v;
        }
        __syncthreads();

#pragma unroll
        for (int k = 0; k < 3; ++k) {
#pragma unroll
            for (int kc = 0; kc < 4; ++kc) {
                const int cil = kc * 4 + (hi ? 2 : 0);
                v2f a0 = *(const v2f*)&Ws[k][(mg << 5) + lane16][cil];
                v2f a1 = *(const v2f*)&Ws[k][(mg << 5) + 16 + lane16][cil];
                v2f b0 = *(const v2f*)&Xs[(ng << 5) + lane16 + k][cil];
                v2f b1 = *(const v2f*)&Xs[(ng << 5) + 16 + lane16 + k][cil];
                acc00 = __builtin_amdgcn_wmma_f32_16x16x4_f32(
                    false, a0, false, b0, (short)0, acc00, false, false);
                acc01 = __builtin_amdgcn_wmma_f32_16x16x4_f32(
                    false, a0, false, b1, (short)0, acc01, false, false);
                acc10 = __builtin_amdgcn_wmma_f32_16x16x4_f32(
                    false, a1, false, b0, (short)0, acc10, false, false);
                acc11 = __builtin_amdgcn_wmma_f32_16x16x4_f32(
                    false, a1, false, b1, (short)0, acc11, false, false);
            }
        }
        __syncthreads();
    }

    const int co_w0 = co0 + (mg << 5);
    const int pos0  = l0 + (ng << 5);
    epilogue_tile<MODE>(acc00, b, co_w0,      pos0,      in, w, w_cin, cout_total,
                        xin, stats, outp, lane16, hi);
    epilogue_tile<MODE>(acc01, b, co_w0,      pos0 + 16, in, w, w_cin, cout_total,
                        xin, stats, outp, lane16, hi);
    epilogue_tile<MODE>(acc10, b, co_w0 + 16, pos0,      in, w, w_cin, cout_total,
                        xin, stats, outp, lane16, hi);
    epilogue_tile<MODE>(acc11, b, co_w0 + 16, pos0 + 16, in, w, w_cin, cout_total,
                        xin, stats, outp, lane16, hi);
}

// ---------------------------------------------------------------------------
// Head: out[b,it,cc,l] = conv1d(H2, h3_w), Cout=2 (tiny FLOPs -> VALU)
// ---------------------------------------------------------------------------
__global__ __launch_bounds__(256)
void h3_kernel(const float* __restrict__ h2buf, const float* __restrict__ w3,
               float* __restrict__ outp, int it) {
    int idx = blockIdx.x * blockDim.x + threadIdx.x;
    if (idx >= NB * 2 * LEN) return;
    int l  = idx & (LEN - 1);
    int cc = (idx >> 9) & 1;
    int b  = idx >> 10;
    float s = 0.f;
    for (int ci = 0; ci < WIDTH / 2; ++ci) {
        const float* hp = h2buf + ((size_t)b * (WIDTH / 2) + ci) * LEN;
        const float* wp = w3 + (cc * (WIDTH / 2) + ci) * 3;
        if (l > 0)       s += wp[0] * hp[l - 1];
        s += wp[1] * hp[l];
        if (l < LEN - 1) s += wp[2] * hp[l + 1];
    }
    outp[(((size_t)b * NITER + it) * 2 + cc) * LEN + l] = s;
}

// ---------------------------------------------------------------------------
extern "C" void kernel_launch(void* const* d_in, const int* in_sizes, int n_in,
                              void* d_out, int out_size, void* d_ws, size_t ws_size,
                              hipStream_t stream) {
    const float* x      = (const float*)d_in[0];
    // d_in[1]: iters_to_do == 16 (fixed by setup; device scalar not host-readable
    // under graph capture)
    const float* proj_w = (const float*)d_in[2];
    const float* op_w   = (const float*)d_in[3];
    const float* h1_w   = (const float*)d_in[4];
    const float* h2_w   = (const float*)d_in[5];
    const float* h3_w   = (const float*)d_in[6];
    float* out = (float*)d_out;

    float* wsf = (float*)d_ws;
    const size_t ACT = (size_t)NB * WIDTH * LEN;          // 6,291,456 floats
    float* bufA  = wsf;
    float* bufB  = bufA + ACT;
    float* H1    = bufB + ACT;
    float* H2    = H1 + ACT;                              // [B,192,L]
    float* stats = H2 + (size_t)NB * (WIDTH / 2) * LEN;   // 64 floats

    proj_kernel<<<(NB * WIDTH * LEN + 255) / 256, 256, 0, stream>>>(x, proj_w, bufA);

    float* cur = bufA;
    float* nxt = bufB;
    for (int it = 0; it < NITER; ++it) {
        ln_stats_kernel<<<NB, 256, 0, stream>>>(cur, stats);
        conv_gemm_kernel<0><<<dim3(NB * 4, WIDTH / 64), 256, 0, stream>>>(
            cur, op_w, WIDTH + 1, WIDTH, x, stats, nxt);
        conv_gemm_kernel<1><<<dim3(NB * 4, WIDTH / 64), 256, 0, stream>>>(
            nxt, h1_w, WIDTH, WIDTH, nullptr, nullptr, H1);
        conv_gemm_kernel<1><<<dim3(NB * 4, (WIDTH / 2) / 64), 256, 0, stream>>>(
            H1, h2_w, WIDTH, WIDTH / 2, nullptr, nullptr, H2);
        h3_kernel<<<(NB * 2 * LEN + 255) / 256, 256, 0, stream>>>(H2, h3_w, out, it);
        float* t = cur; cur = nxt; nxt = t;
    }
}